// GINELayer_4638564679686
// MI455X (gfx1250) — compile-verified
//
#include <hip/hip_runtime.h>
#include <hip/hip_bf16.h>
#include <stdint.h>

// CDNA5 wave32 WMMA vector types
typedef __attribute__((ext_vector_type(2))) float v2f;
typedef __attribute__((ext_vector_type(8))) float v8f;

#define D 64
#define TILES_PER_BLOCK 4   // 4 waves of 32 -> 128 threads/block

// ---------------------------------------------------------------------------
// Kernel 1: out = x   (initializes accumulator with the (1+eps)*x term, eps=0)
// ---------------------------------------------------------------------------
__global__ __launch_bounds__(256) void gine_init_kernel(
    const float* __restrict__ x, float* __restrict__ out, int n4) {
  int i = blockIdx.x * blockDim.x + threadIdx.x;
  if (i < n4) {
    reinterpret_cast<float4*>(out)[i] =
        reinterpret_cast<const float4*>(x)[i];
  }
}

// ---------------------------------------------------------------------------
// Kernel 2: per-edge scatter:  out[dst] += relu(x[src] + edge_attr[e])
// 16 lanes per edge, float4 per lane (64 floats = one D-row per 16 lanes).
// f32 atomics resolve in L2 (agg buffer is 25.6 MB, L2-resident on MI455X).
// ---------------------------------------------------------------------------
__global__ __launch_bounds__(256) void gine_edge_kernel(
    const float* __restrict__ x,
    const long long* __restrict__ edge_index,  // [2, E] int64
    const float* __restrict__ edge_attr,       // [E, 64]
    float* __restrict__ agg,                   // [N, 64] (== d_out)
    int n_edges) {
  int t = blockIdx.x * 256 + threadIdx.x;
  int e = t >> 4;                 // edge id
  int c = (t & 15) * 4;           // column base within D
  if (e >= n_edges) return;

  long long s = edge_index[e];
  long long d = edge_index[n_edges + e];

  float4 ev = *reinterpret_cast<const float4*>(edge_attr + (size_t)e * D + c);
  float4 xv = *reinterpret_cast<const float4*>(x + (size_t)s * D + c);

  float4 m;
  m.x = fmaxf(xv.x + ev.x, 0.0f);
  m.y = fmaxf(xv.y + ev.y, 0.0f);
  m.z = fmaxf(xv.z + ev.z, 0.0f);
  m.w = fmaxf(xv.w + ev.w, 0.0f);

  float* dp = agg + (size_t)d * D + c;
  atomicAdd(dp + 0, m.x);   // global_atomic_add_f32 (no return)
  atomicAdd(dp + 1, m.y);
  atomicAdd(dp + 2, m.z);
  atomicAdd(dp + 3, m.w);
}

// ---------------------------------------------------------------------------
// Kernel 3: per-node MLP via V_WMMA_F32_16X16X4_F32 (fp32, matches reference).
// Each wave32 owns a 16-node tile:  out16 = relu(h16@W1 + b1) @ W2 + b2,
// reading h16 from d_out and writing the result back to the same rows.
// Weights are staged transposed in LDS so B fragments are contiguous float2.
// ---------------------------------------------------------------------------
__global__ __launch_bounds__(128) void gine_mlp_kernel(
    const float* __restrict__ h_in,   // [N, 64] = x + agg  (== d_out)
    const float* __restrict__ W1, const float* __restrict__ b1,
    const float* __restrict__ W2, const float* __restrict__ b2,
    float* __restrict__ out,          // [N, 64]            (== d_out)
    int n_tiles) {
  __shared__ float sW1[D * D];                    // transposed: sW1[n*64+k]
  __shared__ float sW2[D * D];                    // transposed: sW2[n*64+k]
  __shared__ float sB1[D];
  __shared__ float sB2[D];
  __shared__ float sH[TILES_PER_BLOCK][16 * D];   // per-wave intermediate

  const int tid = threadIdx.x;

  // Stage weights (transposed) + biases into LDS cooperatively.
  for (int i = tid; i < D * D; i += 128) {
    int k = i >> 6, n = i & 63;
    sW1[n * D + k] = W1[i];
    sW2[n * D + k] = W2[i];
  }
  if (tid < D) { sB1[tid] = b1[tid]; sB2[tid] = b2[tid]; }
  __syncthreads();

  const int wave = tid >> 5;
  const int lane = tid & 31;
  const int tile = blockIdx.x * TILES_PER_BLOCK + wave;
  if (tile >= n_tiles) return;   // uniform per wave -> EXEC all-1s for WMMA

  const int row  = lane & 15;    // A-matrix row owned by this lane
  const int half = lane >> 4;    // 0: K={0,1} / M rows 0-7 ; 1: K={2,3} / M rows 8-15
  const int col  = lane & 15;    // B/C/D column owned by this lane

  float* myH = &sH[wave][0];

  // ---- load A fragments for GEMM1 from global (row-major h) ----
  const float* arow = h_in + ((size_t)tile * 16 + row) * D + 2 * half;
  v2f a1[16];
#pragma unroll
  for (int kc = 0; kc < 16; ++kc)
    a1[kc] = *reinterpret_cast<const v2f*>(arow + 4 * kc);

  // ---- GEMM1: H = relu(A@W1 + b1), accumulate per 16x16 N-tile ----
#pragma unroll
  for (int n = 0; n < 4; ++n) {
    float bias = sB1[n * 16 + col];
    v8f acc;
#pragma unroll
    for (int r = 0; r < 8; ++r) acc[r] = bias;
#pragma unroll
    for (int kc = 0; kc < 16; ++kc) {
      v2f bfrag = *reinterpret_cast<const v2f*>(
          &sW1[(n * 16 + col) * D + 4 * kc + 2 * half]);
      acc = __builtin_amdgcn_wmma_f32_16x16x4_f32(
          /*neg_a=*/false, a1[kc], /*neg_b=*/false, bfrag,
          /*c_mod=*/(short)0, acc, /*reuse_a=*/false, /*reuse_b=*/false);
    }
    // relu + park in LDS (C layout: M = r + 8*half)
#pragma unroll
    for (int r = 0; r < 8; ++r) {
      float v = acc[r];
      myH[(r + half * 8) * D + n * 16 + col] = (v > 0.0f) ? v : 0.0f;
    }
  }

  // ---- reload H in A-matrix layout (DS ops are in-order per wave) ----
  v2f a2[16];
#pragma unroll
  for (int kc = 0; kc < 16; ++kc)
    a2[kc] = *reinterpret_cast<const v2f*>(&myH[row * D + 4 * kc + 2 * half]);

  // ---- GEMM2: out = H@W2 + b2 ----
#pragma unroll
  for (int n = 0; n < 4; ++n) {
    float bias = sB2[n * 16 + col];
    v8f acc;
#pragma unroll
    for (int r = 0; r < 8; ++r) acc[r] = bias;
#pragma unroll
    for (int kc = 0; kc < 16; ++kc) {
      v2f bfrag = *reinterpret_cast<const v2f*>(
          &sW2[(n * 16 + col) * D + 4 * kc + 2 * half]);
      acc = __builtin_amdgcn_wmma_f32_16x16x4_f32(
          false, a2[kc], false, bfrag, (short)0, acc, false, false);
    }
#pragma unroll
    for (int r = 0; r < 8; ++r) {
      out[((size_t)tile * 16 + r + half * 8) * D + n * 16 + col] = acc[r];
    }
  }
}

// ---------------------------------------------------------------------------
// Launch: inputs in setup_inputs() order:
//   0:x [N,64] f32   1:edge_index [2,E] i64   2:edge_attr [E,64] f32
//   3:W1 [64,64]     4:b1 [64]    5:W2 [64,64]    6:b2 [64]
// d_out: [N,64] f32 — used as the aggregation accumulator, then rewritten
// in-place by the MLP kernel (each wave reads its own rows before writing).
// ---------------------------------------------------------------------------
extern "C" void kernel_launch(void* const* d_in, const int* in_sizes, int n_in,
                              void* d_out, int out_size, void* d_ws, size_t ws_size,
                              hipStream_t stream) {
  const float*     x  = (const float*)d_in[0];
  const long long* ei = (const long long*)d_in[1];   // int64 edge indices
  const float*     ea = (const float*)d_in[2];
  const float*     W1 = (const float*)d_in[3];
  const float*     b1 = (const float*)d_in[4];
  const float*     W2 = (const float*)d_in[5];
  const float*     b2 = (const float*)d_in[6];
  float* out = (float*)d_out;

  const int n_nodes = in_sizes[0] / D;        // 100000
  const int n_edges = in_sizes[2] / D;        // 1600000

  // 1) out = x
  {
    int n4 = (n_nodes * D) / 4;
    int blocks = (n4 + 255) / 256;
    gine_init_kernel<<<blocks, 256, 0, stream>>>(x, out, n4);
  }

  // 2) out[dst] += relu(x[src] + e)   (16 lanes per edge)
  {
    long long threads = (long long)n_edges * 16;
    int blocks = (int)((threads + 255) / 256);
    gine_edge_kernel<<<blocks, 256, 0, stream>>>(x, ei, ea, out, n_edges);
  }

  // 3) out = relu(out@W1 + b1)@W2 + b2   (WMMA fp32, 16 nodes per wave)
  {
    int n_tiles = (n_nodes + 15) / 16;        // 6250
    int blocks = (n_tiles + TILES_PER_BLOCK - 1) / TILES_PER_BLOCK;
    gine_mlp_kernel<<<blocks, 128, 0, stream>>>(out, W1, b1, W2, b2, out, n_tiles);
  }
}